// GeoSGConv_31894427140228
// MI455X (gfx1250) — compile-verified
//
#include <hip/hip_runtime.h>
#include <math.h>

#define N_NODES 100000
#define N_EDGES 3200000
#define NFEAT   256
#define NHID    128
#define NCLASS  16

typedef __attribute__((ext_vector_type(2))) float v2f;
typedef __attribute__((ext_vector_type(8))) float v8f;

// ---------------------------------------------------------------- degree pass
__global__ void k_deg_init(float* __restrict__ deg) {
    int i = blockIdx.x * blockDim.x + threadIdx.x;
    if (i < N_NODES) deg[i] = 1.0f;               // self-loop contributes weight 1
}

__global__ void k_deg_accum(const int* __restrict__ ei, const float* __restrict__ ew,
                            float* __restrict__ deg) {
    int e = blockIdx.x * blockDim.x + threadIdx.x;
    if (e < N_EDGES) atomicAdd(&deg[ei[N_EDGES + e]], ew[e]);   // segment_sum by col
}

__global__ void k_dinv(const float* __restrict__ deg, float* __restrict__ dinv) {
    int i = blockIdx.x * blockDim.x + threadIdx.x;
    if (i < N_NODES) {
        float d = deg[i];
        dinv[i] = (d > 0.0f) ? rsqrtf(d) : 0.0f;
    }
}

// ------------------------------------------------- GEMM1: C[N,128] = X[N,256] @ W1
// grid = 6250 blocks (one 16-row M tile each), 256 threads = 8 waves,
// wave w computes the 16x16 tile at columns [16w, 16w+16). Exact grid => EXEC all-1s.
__global__ void k_gemm1(const float* __restrict__ X, const float* __restrict__ W,
                        float* __restrict__ C) {
    const int wave = threadIdx.x >> 5;
    const int lane = threadIdx.x & 31;
    const int half = lane >> 4;        // lane half selects K sub-pair / C row offset
    const int l    = lane & 15;
    const int m0   = blockIdx.x * 16;
    const int n0   = wave * 16;

    const float* arow = X + (size_t)(m0 + l) * NFEAT + 2 * half;  // A frag base
    const float* bcol = W + (size_t)(2 * half) * NHID + n0 + l;   // B frag base

    v8f acc = {0.f, 0.f, 0.f, 0.f, 0.f, 0.f, 0.f, 0.f};
    for (int k0 = 0; k0 < NFEAT; k0 += 4) {
        // Unconditional speculative prefetch (TH=RT speculative: translation
        // failures are silently dropped, so running past the row end is safe).
        __builtin_prefetch(arow + k0 + 64, 0, 1);                 // global_prefetch_b8
        v2f a = *(const v2f*)(arow + k0);                         // K = k0+2h, k0+2h+1
        v2f b;
        b.x = bcol[(size_t)k0 * NHID];                            // row K = k0+2h
        b.y = bcol[(size_t)(k0 + 1) * NHID];                      // row K = k0+2h+1
        acc = __builtin_amdgcn_wmma_f32_16x16x4_f32(
            /*neg_a=*/false, a, /*neg_b=*/false, b,
            /*c_mod=*/(short)0, acc, /*reuse_a=*/false, /*reuse_b=*/false);
    }
#pragma unroll
    for (int v = 0; v < 8; ++v)
        C[(size_t)(m0 + v + 8 * half) * NHID + n0 + l] = acc[v];
}

// ------------------------------------------------- GEMM2: Z[N,16] = H[N,128] @ W2
// one wave per block (blockDim=32) -> no divergent guard around WMMA.
__global__ void k_gemm2(const float* __restrict__ H, const float* __restrict__ W,
                        float* __restrict__ Z) {
    const int lane = threadIdx.x & 31;
    const int half = lane >> 4;
    const int l    = lane & 15;
    const int m0   = blockIdx.x * 16;

    const float* arow = H + (size_t)(m0 + l) * NHID + 2 * half;
    const float* bcol = W + (size_t)(2 * half) * NCLASS + l;

    v8f acc = {0.f, 0.f, 0.f, 0.f, 0.f, 0.f, 0.f, 0.f};
    for (int k0 = 0; k0 < NHID; k0 += 4) {
        v2f a = *(const v2f*)(arow + k0);
        v2f b;
        b.x = bcol[(size_t)k0 * NCLASS];
        b.y = bcol[(size_t)(k0 + 1) * NCLASS];
        acc = __builtin_amdgcn_wmma_f32_16x16x4_f32(
            false, a, false, b, (short)0, acc, false, false);
    }
#pragma unroll
    for (int v = 0; v < 8; ++v)
        Z[(size_t)(m0 + v + 8 * half) * NCLASS + l] = acc[v];
}

// ------------------------------------------------- layer-1 aggregation (128-dim)
__global__ void k_agg1_init(const float* __restrict__ XW, const float* __restrict__ dinv,
                            float* __restrict__ Hagg) {
    size_t idx  = (size_t)blockIdx.x * blockDim.x + threadIdx.x;   // over N*NHID (exact)
    int    node = (int)(idx >> 7);
    float  di   = dinv[node];
    Hagg[idx]   = di * di * XW[idx];                               // self-loop term
}

// wave per edge; lane handles 4 consecutive features (float4 gather, 4 f32 atomics)
__global__ void k_agg1_edges(const int* __restrict__ ei, const float* __restrict__ ew,
                             const float* __restrict__ dinv, const float* __restrict__ XW,
                             float* __restrict__ Hagg) {
    const int e    = (blockIdx.x * blockDim.x + threadIdx.x) >> 5; // exact grid
    const int lane = threadIdx.x & 31;
    const int r  = ei[e];
    const int c  = ei[N_EDGES + e];
    const float norm = dinv[r] * ew[e] * dinv[c];
    const float4 v = *(const float4*)(XW + (size_t)r * NHID + lane * 4);
    float* dst = Hagg + (size_t)c * NHID + lane * 4;
    atomicAdd(dst + 0, norm * v.x);
    atomicAdd(dst + 1, norm * v.y);
    atomicAdd(dst + 2, norm * v.z);
    atomicAdd(dst + 3, norm * v.w);
}

__global__ void k_bias_relu(float* __restrict__ H, const float* __restrict__ b1) {
    size_t idx = (size_t)blockIdx.x * blockDim.x + threadIdx.x;    // over N*NHID (exact)
    float  h   = H[idx] + b1[idx & (NHID - 1)];
    H[idx] = fmaxf(h, 0.0f);
}

// ------------------------------------------------- layer-2 aggregation (16-dim)
__global__ void k_agg2_init(const float* __restrict__ Z, const float* __restrict__ dinv,
                            float* __restrict__ out) {
    size_t idx  = (size_t)blockIdx.x * blockDim.x + threadIdx.x;   // over N*NCLASS (exact)
    int    node = (int)(idx >> 4);
    float  di   = dinv[node];
    out[idx] = di * di * Z[idx];                                   // self-loop term
}

// 16 threads per edge, one class each
__global__ void k_agg2_edges(const int* __restrict__ ei, const float* __restrict__ ew,
                             const float* __restrict__ dinv, const float* __restrict__ Z,
                             float* __restrict__ out) {
    const size_t tid = (size_t)blockIdx.x * blockDim.x + threadIdx.x;  // exact grid
    const int e = (int)(tid >> 4);
    const int c = (int)(tid & 15);
    const int r  = ei[e];
    const int cc = ei[N_EDGES + e];
    const float norm = dinv[r] * ew[e] * dinv[cc];
    atomicAdd(&out[(size_t)cc * NCLASS + c], norm * Z[(size_t)r * NCLASS + c]);
}

// ------------------------------------------------- bias + log_softmax (in place)
__global__ void k_logsoftmax(float* __restrict__ out, const float* __restrict__ b2) {
    int i = blockIdx.x * blockDim.x + threadIdx.x;
    if (i >= N_NODES) return;
    float* row = out + (size_t)i * NCLASS;
    float v[NCLASS];
    float m = -3.402823466e38f;
#pragma unroll
    for (int c = 0; c < NCLASS; ++c) {
        v[c] = row[c] + b2[c];
        m = fmaxf(m, v[c]);
    }
    float s = 0.0f;
#pragma unroll
    for (int c = 0; c < NCLASS; ++c) s += expf(v[c] - m);
    const float lse = m + logf(s);
#pragma unroll
    for (int c = 0; c < NCLASS; ++c) row[c] = v[c] - lse;
}

// ---------------------------------------------------------------- launcher
extern "C" void kernel_launch(void* const* d_in, const int* in_sizes, int n_in,
                              void* d_out, int out_size, void* d_ws, size_t ws_size,
                              hipStream_t stream) {
    const float* X  = (const float*)d_in[0];
    const int*   ei = (const int*)  d_in[1];
    const float* ew = (const float*)d_in[2];
    const float* W1 = (const float*)d_in[3];
    const float* b1 = (const float*)d_in[4];
    const float* W2 = (const float*)d_in[5];
    const float* b2 = (const float*)d_in[6];
    float* out = (float*)d_out;

    float* ws   = (float*)d_ws;
    float* deg  = ws;                                  // N
    float* dinv = ws + N_NODES;                        // N
    float* buf1 = ws + 2 * (size_t)N_NODES;            // N*NHID : XW1, later Z
    float* buf2 = buf1 + (size_t)N_NODES * NHID;       // N*NHID : Hagg / H

    // normalization
    k_deg_init <<<(N_NODES + 255) / 256, 256, 0, stream>>>(deg);
    k_deg_accum<<<N_EDGES / 256,         256, 0, stream>>>(ei, ew, deg);
    k_dinv     <<<(N_NODES + 255) / 256, 256, 0, stream>>>(deg, dinv);

    // layer 1: A(X W1) + b1, relu
    k_gemm1     <<<N_NODES / 16,               256, 0, stream>>>(X, W1, buf1);
    k_agg1_init <<<(size_t)N_NODES * NHID / 256, 256, 0, stream>>>(buf1, dinv, buf2);
    k_agg1_edges<<<(size_t)N_EDGES * 32 / 256,   256, 0, stream>>>(ei, ew, dinv, buf1, buf2);
    k_bias_relu <<<(size_t)N_NODES * NHID / 256, 256, 0, stream>>>(buf2, b1);

    // layer 2: A(H W2) + b2, log_softmax  (accumulate directly into d_out)
    k_gemm2     <<<N_NODES / 16,                 32, 0, stream>>>(buf2, W2, buf1);
    k_agg2_init <<<(size_t)N_NODES * NCLASS / 256, 256, 0, stream>>>(buf1, dinv, out);
    k_agg2_edges<<<(size_t)N_EDGES * 16 / 256,     256, 0, stream>>>(ei, ew, dinv, buf1, out);
    k_logsoftmax<<<(N_NODES + 255) / 256,          256, 0, stream>>>(out, b2);
}